// l_SAGE_58308476011187
// MI455X (gfx1250) — compile-verified
//
#include <hip/hip_runtime.h>
#include <hip/hip_bf16.h>

typedef __attribute__((ext_vector_type(2))) float v2f;
typedef __attribute__((ext_vector_type(8))) float v8f;

#define D_FEAT 64

// ---------------------------------------------------------------------------
// Kernel 1: zero the scratch accumulators (summed[N*64] ++ count[N]).
// ---------------------------------------------------------------------------
__global__ void sage_zero_kernel(float* __restrict__ p, long long n) {
    long long i = (long long)blockIdx.x * blockDim.x + threadIdx.x;
    if (i < n) p[i] = 0.0f;
}

// ---------------------------------------------------------------------------
// Kernel 2: edge scatter.  Thread t handles edge e = t>>4 and a float4 chunk
// of its 64-float message row.  x is L2-resident (25.6MB << 192MB L2), so the
// gather mostly hits L2; the 82M f32 atomics land in L2 atomic units.
// ---------------------------------------------------------------------------
__global__ void sage_edge_kernel(const int* __restrict__ rows,
                                 const int* __restrict__ cols,
                                 const float* __restrict__ x,
                                 float* __restrict__ summed,
                                 float* __restrict__ count,
                                 int n_edges) {
    long long idx = (long long)blockIdx.x * blockDim.x + threadIdx.x;
    int e = (int)(idx >> 4);
    if (e >= n_edges) return;
    int c4 = ((int)idx & 15) * 4;
    int r = rows[e];
    int c = cols[e];
    const float4 v = *(const float4*)(x + (long long)c * D_FEAT + c4);
    float* dst = summed + (long long)r * D_FEAT + c4;
    atomicAdd(dst + 0, v.x);
    atomicAdd(dst + 1, v.y);
    atomicAdd(dst + 2, v.z);
    atomicAdd(dst + 3, v.w);
    if ((idx & 15) == 0) atomicAdd(count + r, 1.0f);
}

// ---------------------------------------------------------------------------
// Kernel 3: out = (summed/max(count,1)) @ W + x @ Wroot + bias
// One wave32 per 16x16 output tile, f32 WMMA 16x16x4, K=64 -> 16 steps x 2
// matrices = 32 v_wmma per wave.  Mean scaling folded into the A loads,
// bias folded into the store.  Tile assignment is wave-uniform so EXEC stays
// all-ones through the WMMAs (loads clamped, stores masked for remainder).
// ---------------------------------------------------------------------------
__global__ void __launch_bounds__(256)
sage_gemm_kernel(const float* __restrict__ summed,
                 const float* __restrict__ count,
                 const float* __restrict__ x,
                 const float* __restrict__ W,
                 const float* __restrict__ Wr,
                 const float* __restrict__ bias,
                 float* __restrict__ out,
                 int n_nodes) {
    const int wave = (int)((blockIdx.x * blockDim.x + threadIdx.x) >> 5);
    const int lane = threadIdx.x & 31;
    const int tile_n = wave & 3;          // 64/16 = 4 N-tiles
    const int tile_m = wave >> 2;
    const int m0 = tile_m * 16;
    if (m0 >= n_nodes) return;            // wave-uniform exit

    const int half = lane >> 4;           // 0: K pair {0,1}, 1: K pair {2,3}
    const int ln   = lane & 15;
    const int n0   = tile_n * 16;

    // A-operand row for this lane (clamped so EXEC stays all-ones)
    int m = m0 + ln;
    if (m >= n_nodes) m = n_nodes - 1;
    const float cnt = count[m];
    const float inv = 1.0f / fmaxf(cnt, 1.0f);

    const float* __restrict__ srow = summed + (long long)m * D_FEAT;
    const float* __restrict__ xrow = x      + (long long)m * D_FEAT;

    v8f acc = {};
    #pragma unroll
    for (int k = 0; k < D_FEAT; k += 4) {
        const int ka = k + 2 * half;      // this lane's K pair within the step
        v2f a, b;

        // agg @ W  (mean scaling folded into A)
        a.x = srow[ka] * inv;
        a.y = srow[ka + 1] * inv;
        b.x = W[ka * D_FEAT + n0 + ln];
        b.y = W[(ka + 1) * D_FEAT + n0 + ln];
        acc = __builtin_amdgcn_wmma_f32_16x16x4_f32(
            false, a, false, b, (short)0, acc, false, false);

        // x @ Wroot
        a.x = xrow[ka];
        a.y = xrow[ka + 1];
        b.x = Wr[ka * D_FEAT + n0 + ln];
        b.y = Wr[(ka + 1) * D_FEAT + n0 + ln];
        acc = __builtin_amdgcn_wmma_f32_16x16x4_f32(
            false, a, false, b, (short)0, acc, false, false);
    }

    // D layout: acc[r] = D[m0 + r + 8*half][n0 + ln]
    const float bval = bias[n0 + ln];
    #pragma unroll
    for (int r = 0; r < 8; ++r) {
        const int mm = m0 + r + 8 * half;
        if (mm < n_nodes)
            out[(long long)mm * D_FEAT + n0 + ln] = acc[r] + bval;
    }
}

// ---------------------------------------------------------------------------
extern "C" void kernel_launch(void* const* d_in, const int* in_sizes, int n_in,
                              void* d_out, int out_size, void* d_ws, size_t ws_size,
                              hipStream_t stream) {
    const float* x    = (const float*)d_in[0];
    const int*   ei   = (const int*)d_in[1];   // (2, E): rows then cols
    const float* W    = (const float*)d_in[2];
    const float* Wr   = (const float*)d_in[3];
    const float* bias = (const float*)d_in[4];
    float* out = (float*)d_out;

    const int n_nodes = in_sizes[0] / D_FEAT;
    const int n_edges = in_sizes[1] / 2;

    float* summed = (float*)d_ws;                         // N*64 floats
    float* count  = summed + (long long)n_nodes * D_FEAT; // N floats

    // 1) zero accumulators (deterministic per call)
    {
        long long zn = (long long)n_nodes * (D_FEAT + 1);
        int blocks = (int)((zn + 255) / 256);
        sage_zero_kernel<<<blocks, 256, 0, stream>>>(summed, zn);
    }

    // 2) edge scatter-add
    {
        long long work = (long long)n_edges * 16;
        int blocks = (int)((work + 255) / 256);
        sage_edge_kernel<<<blocks, 256, 0, stream>>>(
            ei, ei + n_edges, x, summed, count, n_edges);
    }

    // 3) WMMA GEMM epilogue
    {
        int tiles_m = (n_nodes + 15) / 16;
        long long waves = (long long)tiles_m * 4;
        int blocks = (int)((waves + 7) / 8);   // 8 waves per 256-thread block
        sage_gemm_kernel<<<blocks, 256, 0, stream>>>(
            summed, count, x, W, Wr, bias, out, n_nodes);
    }
}